// SimpleModel_58755152609320
// MI455X (gfx1250) — compile-verified
//
#include <hip/hip_runtime.h>
#include <math.h>

typedef float v2f __attribute__((ext_vector_type(2)));
typedef float v8f __attribute__((ext_vector_type(8)));

#define EMB 300
#define K_STEPS 75   // EMB / 4

// ---------------------------------------------------------------------------
// Kernel 1: dotA[v] = emb[v] . W[0:300], dotB[v] = emb[v] . W[300:600]
// One wave (32 threads) per 16 vocab rows, via V_WMMA_F32_16X16X4_F32.
//   A (16x4 f32): lanes 0-15 hold rows, VGPR0/1 = K {k,k+1}; lanes 16-31 = {k+2,k+3}
//   B (4x16 f32): lane n = output column; only n=0 (W col A) and n=1 (W col B) nonzero
//   D (16x16 f32): VGPR r -> M=r (lanes 0-15) / M=8+r (lanes 16-31), N = lane&15
// ---------------------------------------------------------------------------
__global__ __launch_bounds__(32)
void dot_gemv_wmma(const float* __restrict__ emb, const float* __restrict__ W,
                   float* __restrict__ dotA, float* __restrict__ dotB, int vocab) {
  __shared__ float ldsA[16 * EMB];   // 19200 B; row stride 1200 B (16B aligned)
  __shared__ float ldsW[2 * EMB];    // 2400 B

  const int lane = threadIdx.x;
  const long row_base = (long)blockIdx.x * 16;

  if (row_base + 16 <= vocab) {
    // Tile rows are consecutive -> the 16x300 tile is one contiguous 19200 B
    // region.  Flat float4 copy: clean global_load_b128 / ds_store_b128 pairs.
    const float4* __restrict__ src = (const float4*)(emb + row_base * EMB);
    float4* __restrict__ dst4 = (float4*)ldsA;
    for (int idx = lane; idx < 16 * K_STEPS; idx += 32) {
      dst4[idx] = src[idx];
    }
  } else {
    // Ragged tail tile (not hit when vocab % 16 == 0): clamped per-row copy.
    for (int idx = lane; idx < 16 * K_STEPS; idx += 32) {
      int row = idx / K_STEPS;
      int c4  = idx % K_STEPS;
      long grow = row_base + row;
      if (grow >= vocab) grow = vocab - 1;
      ((float4*)(ldsA + row * EMB))[c4] = ((const float4*)(emb + grow * EMB))[c4];
    }
  }
  // W is 600 floats = 150 float4
  for (int idx = lane; idx < 2 * K_STEPS; idx += 32) {
    ((float4*)ldsW)[idx] = ((const float4*)W)[idx];
  }
  __syncthreads();

  const int n  = lane & 15;   // output column (only 0,1 meaningful) / A row
  const int hi = lane >> 4;   // K-half selector
  const bool has_b = (n < 2);

  const float* aptr = ldsA + n * EMB + 2 * hi;
  const float* bptr = ldsW + (has_b ? n * EMB : 0) + 2 * hi;

  v8f c = {0.f, 0.f, 0.f, 0.f, 0.f, 0.f, 0.f, 0.f};
  #pragma unroll
  for (int k0 = 0; k0 < EMB; k0 += 4) {
    v2f a, b;
    a.x = aptr[k0];
    a.y = aptr[k0 + 1];
    b.x = has_b ? bptr[k0]     : 0.f;
    b.y = has_b ? bptr[k0 + 1] : 0.f;
    c = __builtin_amdgcn_wmma_f32_16x16x4_f32(
        /*neg_a=*/false, a, /*neg_b=*/false, b,
        /*c_mod=*/(short)0, c, /*reuse_a=*/false, /*reuse_b=*/false);
  }

  // Write out columns N=0 (dotA) and N=1 (dotB).
  if (has_b) {
    float* dst = (n == 0) ? dotA : dotB;
    long r0 = row_base + hi * 8;
    #pragma unroll
    for (int r = 0; r < 8; ++r) {
      long row = r0 + r;
      if (row < vocab) dst[row] = c[r];
    }
  }
}

// ---------------------------------------------------------------------------
// Kernel 2a: zero-init segment bounds (empty segments -> count 0 -> mean 0)
// ---------------------------------------------------------------------------
__global__ void init_bounds(int* __restrict__ start, int* __restrict__ end, int S) {
  int s = blockIdx.x * blockDim.x + threadIdx.x;
  if (s < S) { start[s] = 0; end[s] = 0; }
}

// ---------------------------------------------------------------------------
// Kernel 2b: segment_ids are sorted -> boundary detection, plain stores only
// (deterministic: exactly one writer per written element).
// ---------------------------------------------------------------------------
__global__ void find_bounds(const int* __restrict__ seg,
                            int* __restrict__ start, int* __restrict__ end, int T) {
  int t = blockIdx.x * blockDim.x + threadIdx.x;
  if (t >= T) return;
  int s = seg[t];
  if (t == 0     || seg[t - 1] != s) start[s] = t;
  if (t == T - 1 || seg[t + 1] != s) end[s]   = t + 1;
}

// ---------------------------------------------------------------------------
// Kernel 3: one wave32 per sentence pair.
//   logit[p] = sum(dotA[tok], seg 2p)/c0 + sum(dotB[tok], seg 2p+1)/c1 + bias
//   out[p]   = sigmoid(logit)
// Fixed-order xor-tree reduction -> deterministic across replays.
// ---------------------------------------------------------------------------
__global__ __launch_bounds__(128)
void pair_reduce(const int* __restrict__ tok,
                 const int* __restrict__ start, const int* __restrict__ end,
                 const float* __restrict__ dotA, const float* __restrict__ dotB,
                 const float* __restrict__ bias,
                 float* __restrict__ out, int npairs) {
  const int wavesPerBlock = blockDim.x >> 5;
  const int pair = blockIdx.x * wavesPerBlock + (threadIdx.x >> 5);
  const int lane = threadIdx.x & 31;
  if (pair >= npairs) return;

  float logit = bias[0];
  #pragma unroll
  for (int h = 0; h < 2; ++h) {
    const int s = 2 * pair + h;
    const int s0 = start[s], e0 = end[s];
    const float* __restrict__ dv = h ? dotB : dotA;  // even seg -> W[:300], odd -> W[300:]
    float sum = 0.f;
    for (int t = s0 + lane; t < e0; t += 32) sum += dv[tok[t]];
    #pragma unroll
    for (int m = 16; m >= 1; m >>= 1) sum += __shfl_xor(sum, m, 32);
    const int cnt = e0 - s0;                         // 0 for empty segments
    logit += sum / fmaxf((float)cnt, 1.0f);
  }
  if (lane == 0) out[pair] = 1.0f / (1.0f + __expf(-logit));
}

// ---------------------------------------------------------------------------
extern "C" void kernel_launch(void* const* d_in, const int* in_sizes, int n_in,
                              void* d_out, int out_size, void* d_ws, size_t ws_size,
                              hipStream_t stream) {
  const float* emb = (const float*)d_in[0];   // [VOCAB, 300]
  const float* W   = (const float*)d_in[1];   // [600, 1]
  const float* b   = (const float*)d_in[2];   // [1]
  const int*   tok = (const int*)d_in[3];     // [T]
  const int*   seg = (const int*)d_in[4];     // [T] sorted
  // d_in[5] = num_segments scalar (device); derive host-side from out_size instead.

  const int T      = in_sizes[3];
  const int vocab  = in_sizes[0] / EMB;       // 100000
  const int npairs = out_size;                // 16384
  const int S      = npairs * 2;              // 32768

  // Workspace layout: dotA[vocab], dotB[vocab], start[S], end[S]
  float* dotA  = (float*)d_ws;
  float* dotB  = dotA + vocab;
  int*   start = (int*)(dotB + vocab);
  int*   end   = start + S;

  const int tiles = (vocab + 15) / 16;        // 6250 waves
  dot_gemv_wmma<<<tiles, 32, 0, stream>>>(emb, W, dotA, dotB, vocab);

  init_bounds<<<(S + 255) / 256, 256, 0, stream>>>(start, end, S);
  find_bounds<<<(T + 255) / 256, 256, 0, stream>>>(seg, start, end, T);

  const int wavesPerBlock = 4;                // 128 threads
  const int blocks = (npairs + wavesPerBlock - 1) / wavesPerBlock;
  pair_reduce<<<blocks, wavesPerBlock * 32, 0, stream>>>(
      tok, start, end, dotA, dotB, b, (float*)d_out, npairs);
}